// ImageAverage_34462817583170
// MI455X (gfx1250) — compile-verified
//
#include <hip/hip_runtime.h>

// ---------------------------------------------------------------------------
// Segmented mean over sorted image ids, MI455X (gfx1250).
//
// Bandwidth-bound: 1.07 GB of x + 32 MB of ids ~= 47 us at 23.3 TB/s.
// The segmented sum is done with V_WMMA_F32_16X16X4_F32:
//   A(16x4)  = one-hot indicator of (id[row] - window_base)   [exact 0/1]
//   B(4x16)  = x tile (4 rows x 16 dims)
//   C/D      = 16 segments x 16 dims f32 accumulator
// Sorted ids => a 16-segment window covers a wave's whole 256-row chunk with
// overwhelming probability, so global f32 atomics (zero-skipped) fire ~once
// per wave. Indicator products are exact, so accumulation is exact f32 adds.
//
// Round-1 fixes: (a) all 4 group ids fetched with one b128 load instead of
// three b32 loads; (b) window base / overflow check forced wave-uniform via
// readfirstlane so the flush branch is scalar (s_cmp/s_cbranch) and EXEC
// stays all-ones around the WMMAs as the ISA requires.
// ---------------------------------------------------------------------------

typedef __attribute__((ext_vector_type(2))) float v2f;
typedef __attribute__((ext_vector_type(8))) float v8f;

#define N_REFLNS 4194304
#define N_DIM    64
#define N_IMAGES 1024

#define SUM_TPB         256
#define SUM_BLOCKS      2048
#define WAVES           (SUM_BLOCKS * (SUM_TPB / 32))  // 16384
#define ROWS_PER_WAVE   (N_REFLNS / WAVES)             // 256
#define GROUPS_PER_WAVE (ROWS_PER_WAVE / 4)            // 64 groups of K=4 rows

// Flush the 16(segments) x 64(dims) accumulator window to global sums.
// C/D layout (ISA 7.12.2): lane l, VGPR j holds (M = j + 8*(l>=16), N = l&15).
// Zero-skip: untouched window rows cost nothing.
__device__ __forceinline__ void flush_acc(float* __restrict__ sums, int wbase,
                                          int lane, v8f& a0, v8f& a1,
                                          v8f& a2, v8f& a3) {
  const int half = lane >> 4;
  const int n    = lane & 15;
#pragma unroll
  for (int j = 0; j < 8; ++j) {
    const int seg = wbase + j + 8 * half;
    if (seg < N_IMAGES) {
      float* base = sums + seg * N_DIM + n;
      float v;
      v = a0[j]; if (v != 0.0f) atomicAdd(base + 0,  v);
      v = a1[j]; if (v != 0.0f) atomicAdd(base + 16, v);
      v = a2[j]; if (v != 0.0f) atomicAdd(base + 32, v);
      v = a3[j]; if (v != 0.0f) atomicAdd(base + 48, v);
    }
  }
  const v8f z = {};
  a0 = z; a1 = z; a2 = z; a3 = z;
}

__global__ __launch_bounds__(SUM_TPB) void seg_sum_wmma(
    const float* __restrict__ x, const int* __restrict__ id,
    float* __restrict__ sums) {
  const int lane = threadIdx.x & 31;
  const int wid  = blockIdx.x * (SUM_TPB / 32) + (threadIdx.x >> 5);
  const long r_begin = (long)wid * ROWS_PER_WAVE;

  // A and B share the same per-lane K mapping (ISA 7.12.2, 32-bit 16x4):
  // lane l, VGPR j -> K = 2*(l>=16) + j ; M/N = l & 15.
  const int half = lane >> 4;
  const int m    = lane & 15;
  const int k0   = half * 2;

  v8f acc0 = {}, acc1 = {}, acc2 = {}, acc3 = {};
  // Wave-uniform window base, kept in an SGPR.
  int wbase = __builtin_amdgcn_readfirstlane(id[r_begin]);

  for (int g = 0; g < GROUPS_PER_WAVE; ++g) {
    const long r0 = r_begin + 4l * g;

    // All 4 group ids in one 16B-aligned b128 load (r0 % 4 == 0).
    const int4 ids = *(const int4*)(id + r0);

    // ids sorted -> group max is ids.w. Force scalar: uniform s_cmp branch.
    const int idmax = __builtin_amdgcn_readfirstlane(ids.w);
    if (idmax - wbase > 15) {  // scalar branch: EXEC untouched
      flush_acc(sums, wbase, lane, acc0, acc1, acc2, acc3);
      wbase = __builtin_amdgcn_readfirstlane(ids.x);
    }

    const int ida = half ? ids.z : ids.x;  // id of row r0 + k0
    const int idb = half ? ids.w : ids.y;  // id of row r0 + k0 + 1
    v2f a;
    a[0] = (ida - wbase == m) ? 1.0f : 0.0f;
    a[1] = (idb - wbase == m) ? 1.0f : 0.0f;

    // B: lane reads x[r0+k][tile*16 + m]; each half-wave is a 64B run.
    const float* xa = x + (r0 + k0) * (long)N_DIM + m;
    const float* xb = xa + N_DIM;

    v2f b;
    b[0] = xa[0];  b[1] = xb[0];
    acc0 = __builtin_amdgcn_wmma_f32_16x16x4_f32(false, a, false, b, (short)0,
                                                 acc0, false, false);
    b[0] = xa[16]; b[1] = xb[16];
    acc1 = __builtin_amdgcn_wmma_f32_16x16x4_f32(false, a, false, b, (short)0,
                                                 acc1, false, false);
    b[0] = xa[32]; b[1] = xb[32];
    acc2 = __builtin_amdgcn_wmma_f32_16x16x4_f32(false, a, false, b, (short)0,
                                                 acc2, false, false);
    b[0] = xa[48]; b[1] = xb[48];
    acc3 = __builtin_amdgcn_wmma_f32_16x16x4_f32(false, a, false, b, (short)0,
                                                 acc3, false, false);
  }
  flush_acc(sums, wbase, lane, acc0, acc1, acc2, acc3);
}

// Fused: copy ids to output (bit-exact) + segment start/end detection on the
// sorted id stream. Empty segments keep start=end=0 -> count 0.
__global__ __launch_bounds__(256) void bounds_copy(
    const int* __restrict__ id, int* __restrict__ out_id,
    int* __restrict__ starts, int* __restrict__ ends) {
  const int i = blockIdx.x * blockDim.x + threadIdx.x;
  if (i >= N_REFLNS) return;
  const int v = id[i];
  out_id[i] = v;
  if (i == 0) starts[v] = 0;
  if (i == N_REFLNS - 1) {
    ends[v] = N_REFLNS;
  } else {
    const int nx = id[i + 1];  // L0/L2 hit (neighbor of own load)
    if (nx != v) { ends[v] = i + 1; starts[nx] = i + 1; }
  }
}

__global__ __launch_bounds__(N_DIM) void finalize(
    const float* __restrict__ sums, const int* __restrict__ starts,
    const int* __restrict__ ends, float* __restrict__ out_avg,
    int* __restrict__ out_counts) {
  const int s = blockIdx.x;
  const int d = threadIdx.x;
  const int cnt = ends[s] - starts[s];
  if (d == 0) out_counts[s] = cnt;
  const float denom = (float)(cnt > 0 ? cnt : 1);
  out_avg[s * N_DIM + d] = sums[s * N_DIM + d] / denom;
}

extern "C" void kernel_launch(void* const* d_in, const int* in_sizes, int n_in,
                              void* d_out, int out_size, void* d_ws,
                              size_t ws_size, hipStream_t stream) {
  (void)in_sizes; (void)n_in; (void)out_size; (void)ws_size;

  const float* x  = (const float*)d_in[0];
  const int*   id = (const int*)d_in[1];
  // d_in[2] = n_images scalar; fixed at N_IMAGES compile-time.

  // Output layout (tuple, flat, in return order):
  //   [0 .. 65535]                      averaged  (f32)
  //   [65536 .. 65536+4194304)          image_id  (i32 bits)
  //   [65536+4194304 .. +1024)          counts    (i32 bits)
  float* out_avg    = (float*)d_out;
  int*   out_id     = (int*)((float*)d_out + (size_t)N_IMAGES * N_DIM);
  int*   out_counts = out_id + (size_t)N_REFLNS;

  // Workspace: sums (256 KB) + starts/ends (8 KB); zero every call.
  float* sums   = (float*)d_ws;
  int*   starts = (int*)((char*)d_ws + (size_t)N_IMAGES * N_DIM * sizeof(float));
  int*   ends   = starts + N_IMAGES;
  hipMemsetAsync(d_ws, 0,
                 (size_t)N_IMAGES * N_DIM * sizeof(float) +
                     2u * N_IMAGES * sizeof(int),
                 stream);

  seg_sum_wmma<<<SUM_BLOCKS, SUM_TPB, 0, stream>>>(x, id, sums);
  bounds_copy<<<N_REFLNS / 256, 256, 0, stream>>>(id, out_id, starts, ends);
  finalize<<<N_IMAGES, N_DIM, 0, stream>>>(sums, starts, ends, out_avg,
                                           out_counts);
}